// dgl_gin_73529840107896
// MI455X (gfx1250) — compile-verified
//
#include <hip/hip_runtime.h>

// GIN 2-layer network, gates degenerate to 1.0 (softmax over size-1 axis), so:
//   x1  = (0.9*(f + A1 f) + 0.8*(f + A2 f)) @ W1 + 1.7*b1
//   out = (0.9*(x1 + A1 x1) + 0.8*(x1 + A2 x1)) @ W2 + 1.7*b2
// Edge aggregation (4x E=600k edges * 128 f32) dominates: ~2.5 GB of RMW, all
// L2-resident (buffers ~77MB << 192MB L2). GEMMs are tiny -> fp32 WMMA
// (v_wmma_f32_16x16x4_f32) keeps full precision at negligible cost.

typedef __attribute__((ext_vector_type(2))) float v2f;
typedef __attribute__((ext_vector_type(8))) float v8f;

#define D 128  // feature width at both GIN inputs (D_IN == HID == 128)

// --- init: acc0 = acc1 = src (self term of GIN: x + sum_neighbors) ----------
__global__ void init2_kernel(const float4* __restrict__ src,
                             float4* __restrict__ a, float4* __restrict__ b,
                             int n4) {
    int i = blockIdx.x * blockDim.x + threadIdx.x;
    if (i < n4) { float4 v = src[i]; a[i] = v; b[i] = v; }
}

// --- scatter-add: acc[dst[e]] += x[src[e]], one wave per edge ---------------
// 32 lanes cover 128 dims with float4; gather is a coalesced 512B line.
__global__ void scatter_kernel(const float* __restrict__ x,
                               const int* __restrict__ src,
                               const int* __restrict__ dst,
                               float* __restrict__ acc, int E) {
    int tid = blockIdx.x * blockDim.x + threadIdx.x;
    int e = tid >> 5;
    if (e >= E) return;
    int c4 = (tid & 31) << 2;
    int s = src[e];
    int d = dst[e];
    const float4 v = *(const float4*)(x + (size_t)s * D + c4);
    float* p = acc + (size_t)d * D + c4;
    unsafeAtomicAdd(p + 0, v.x);   // global_atomic_add_f32
    unsafeAtomicAdd(p + 1, v.y);
    unsafeAtomicAdd(p + 2, v.z);
    unsafeAtomicAdd(p + 3, v.w);
}

// --- fused combine + GEMM:  out = (s0*in0 + s1*in1) @ W + bscale*bias -------
// One wave computes one 16x16 output tile via 32 chained V_WMMA_F32_16X16X4_F32.
// blockDim.x = (NC/16)*32 waves cover a 16-row x NC-col strip; grid = N/16.
// NC is a template parameter so all weight loads use immediate 24-bit offsets
// (k*NC*4 <= 65KB) instead of v_add_nc_u64 pointer chains.
// A frag (16x4 f32): lanes 0-15 hold K=0,1 ; lanes 16-31 hold K=2,3 (rows M=lane&15)
// B frag (4x16 f32): VGPR0 = row K+2*half, VGPR1 = row K+2*half+1, col = n0+(lane&15)
// C/D (16x16 f32):   VGPR v -> row v (lanes 0-15) / row v+8 (lanes 16-31)
template <int NC>
__global__ void gin_gemm_kernel(const float* __restrict__ in0,
                                const float* __restrict__ in1,
                                float s0, float s1,
                                const float* __restrict__ W,   // [D x NC], row-major
                                const float* __restrict__ bias, float bscale,
                                float* __restrict__ out) {
    int lane = threadIdx.x & 31;
    int wave = threadIdx.x >> 5;
    int n0   = wave * 16;
    int m0   = blockIdx.x * 16;
    int half = lane >> 4;      // 0: K pair {0,1}, 1: K pair {2,3}
    int l15  = lane & 15;

    const float* r0 = in0 + (size_t)(m0 + l15) * D + 2 * half;
    const float* r1 = in1 + (size_t)(m0 + l15) * D + 2 * half;
    const float* wc = W + (size_t)(2 * half) * NC + n0 + l15;

    v8f acc = {};
#pragma unroll
    for (int k = 0; k < D; k += 4) {
        v2f a0 = *(const v2f*)(r0 + k);
        v2f a1 = *(const v2f*)(r1 + k);
        v2f a;
        a[0] = s0 * a0[0] + s1 * a1[0];
        a[1] = s0 * a0[1] + s1 * a1[1];
        v2f b;
        b[0] = wc[k * NC];           // compile-time immediate offsets
        b[1] = wc[(k + 1) * NC];
        // (neg_a, A, neg_b, B, c_mod, C, reuse_a, reuse_b)
        acc = __builtin_amdgcn_wmma_f32_16x16x4_f32(
            false, a, false, b, (short)0, acc, false, false);
    }

    float bv = bscale * bias[n0 + l15];
#pragma unroll
    for (int v = 0; v < 8; ++v) {
        int row = m0 + v + half * 8;
        out[(size_t)row * NC + n0 + l15] = acc[v] + bv;
    }
}

extern "C" void kernel_launch(void* const* d_in, const int* in_sizes, int n_in,
                              void* d_out, int out_size, void* d_ws, size_t ws_size,
                              hipStream_t stream) {
    const float* features = (const float*)d_in[0];
    const int*   src1     = (const int*)d_in[1];
    const int*   dst1     = (const int*)d_in[2];
    const int*   src2     = (const int*)d_in[3];
    const int*   dst2     = (const int*)d_in[4];
    const float* W1       = (const float*)d_in[5];
    const float* b1       = (const float*)d_in[6];
    const float* W2       = (const float*)d_in[7];
    const float* b2       = (const float*)d_in[8];
    // gates (d_in[9..16]) are mathematically inert: softmax over a size-1 axis == 1.0

    const int N = in_sizes[0] / D;   // 50000 (divisible by 16)
    const int E = in_sizes[1];       // 600000

    float* buf0 = (float*)d_ws;                 // accumulator graph 1
    float* buf1 = buf0 + (size_t)N * D;         // accumulator graph 2
    float* buf2 = buf1 + (size_t)N * D;         // x1

    const int n4      = N * D / 4;
    const int cblocks = (n4 + 255) / 256;
    const int sblocks = (E * 32 + 255) / 256;

    // ---- layer 1 ----
    init2_kernel<<<cblocks, 256, 0, stream>>>((const float4*)features,
                                              (float4*)buf0, (float4*)buf1, n4);
    scatter_kernel<<<sblocks, 256, 0, stream>>>(features, src1, dst1, buf0, E);
    scatter_kernel<<<sblocks, 256, 0, stream>>>(features, src2, dst2, buf1, E);
    gin_gemm_kernel<128><<<N / 16, 8 * 32, 0, stream>>>(buf0, buf1, 0.9f, 0.8f,
                                                        W1, b1, 1.7f, buf2);
    // ---- layer 2 ----
    init2_kernel<<<cblocks, 256, 0, stream>>>((const float4*)buf2,
                                              (float4*)buf0, (float4*)buf1, n4);
    scatter_kernel<<<sblocks, 256, 0, stream>>>(buf2, src1, dst1, buf0, E);
    scatter_kernel<<<sblocks, 256, 0, stream>>>(buf2, src2, dst2, buf1, E);
    gin_gemm_kernel<64><<<N / 16, 4 * 32, 0, stream>>>(buf0, buf1, 0.9f, 0.8f,
                                                       W2, b2, 1.7f, (float*)d_out);
}